// PoolToVertices_58480274702421
// MI455X (gfx1250) — compile-verified
//
#include <hip/hip_runtime.h>

// Problem sizes (fixed by the reference)
#define BB 4
#define NN 2048
#define EE 4096
#define DD 256

typedef __attribute__((ext_vector_type(2))) float v2f;
typedef __attribute__((ext_vector_type(4))) float f4;
typedef __attribute__((ext_vector_type(8))) float v8f;

// CDNA5 async global->LDS copy (GV mode: vdst = LDS byte address, vaddr = 64-bit
// global address). Tracked by ASYNCcnt. Generic-pointer low 32 bits of a
// __shared__ object are its LDS byte offset (aperture bits live in [63:32]).
#define ASYNC_LOAD_B128(lds_u32, gptr)                                        \
  asm volatile("global_load_async_to_lds_b128 %0, %1, off"                    \
               :: "v"(lds_u32), "v"(gptr) : "memory")

#define WAIT_ASYNCCNT_0() asm volatile("s_wait_asynccnt 0x0" ::: "memory")

__device__ __forceinline__ unsigned lds_addr(const void* p) {
  return (unsigned)(unsigned long long)p;
}

// ---------------------------------------------------------------------------
// Fused: v_out = v_in + adj^T @ v_in + conEd @ e_in + u  (per batch)
//
// Block = 8 waves = 128(n) x 16(d) macro-tile; wave w owns n-tile w.
// K is processed in chunks of 32, double-buffered in LDS via async-to-LDS.
//   Phase 1 (K = N = 2048, 64 chunks):  A = adj rows (k x 128n), LDS [32][144]
//   Phase 2 (K = E = 4096, 128 chunks): A = conEd rows (128n x k), LDS [128][36]
//   B (both phases): 32 K-rows x 16 d-cols of v_in / e_in, LDS [32][16]
// ---------------------------------------------------------------------------
__global__ __launch_bounds__(256) void fused_pool_wmma(
    const float* __restrict__ v_in, const float* __restrict__ e_in,
    const float* __restrict__ u_in, const float* __restrict__ adj,
    const float* __restrict__ conEd, float* __restrict__ v_out)
{
  __shared__ float sA[2][4608];   // 32*144 (phase1) == 128*36 (phase2)
  __shared__ float sB[2][512];    // 32*16

  const int t    = threadIdx.x;
  const int wave = t >> 5;
  const int lane = t & 31;
  const int half = lane >> 4;     // 0: K pair {0,1}, 1: K pair {2,3}
  const int col  = lane & 15;
  const int w16  = wave << 4;

  const int idx = blockIdx.x;     // 1024 blocks: b(4) x nb(16) x dt(16)
  const int b   = idx >> 8;
  const int nb0 = ((idx >> 4) & 15) << 7;   // macro-tile n origin (128-row)
  const int d0  = (idx & 15) << 4;          // d-tile origin

  const size_t off_adj = (size_t)b * NN * NN;
  const size_t off_v   = (size_t)b * NN * DD;
  const size_t off_con = (size_t)b * NN * EE;
  const size_t off_e   = (size_t)b * EE * DD;

  const int C1 = NN / 32;          // 64 phase-1 chunks
  const int C  = C1 + EE / 32;     // + 128 phase-2 chunks = 192

  // ---- stage chunk c into buffer `buf` (async, no wait here) --------------
  auto stage = [&](int c, int buf) {
    if (c < C1) {
      const int k0 = c << 5;
      // A: adj[k0..k0+32, nb0..nb0+128] -> sA[buf][row*144 + col], 1024 b128 segs
      const float* gA = adj + off_adj + (size_t)k0 * NN + nb0;
      #pragma unroll
      for (int i = 0; i < 4; ++i) {
        const int seg = i * 256 + t;         // 32 rows x 32 segs
        const int row = seg >> 5, s = seg & 31;
        ASYNC_LOAD_B128(lds_addr(&sA[buf][row * 144 + s * 4]),
                        gA + (size_t)row * NN + s * 4);
      }
      if (t < 128) {                          // B: v_in[k0..k0+32, d0..d0+16]
        const int row = t >> 2, s = t & 3;
        ASYNC_LOAD_B128(lds_addr(&sB[buf][row * 16 + s * 4]),
                        v_in + off_v + (size_t)(k0 + row) * DD + d0 + s * 4);
      }
    } else {
      const int k0 = (c - C1) << 5;
      // A: conEd[nb0..nb0+128, k0..k0+32] -> sA[buf][row*36 + k], 1024 b128 segs
      const float* gA = conEd + off_con + (size_t)nb0 * EE + k0;
      #pragma unroll
      for (int i = 0; i < 4; ++i) {
        const int seg = i * 256 + t;         // 128 rows x 8 segs
        const int row = seg >> 3, s = seg & 7;
        ASYNC_LOAD_B128(lds_addr(&sA[buf][row * 36 + s * 4]),
                        gA + (size_t)row * EE + s * 4);
      }
      if (t < 128) {                          // B: e_in[k0..k0+32, d0..d0+16]
        const int row = t >> 2, s = t & 3;
        ASYNC_LOAD_B128(lds_addr(&sB[buf][row * 16 + s * 4]),
                        e_in + off_e + (size_t)(k0 + row) * DD + d0 + s * 4);
      }
    }
  };

  v8f acc = {};

  // ---- pipeline: stage(c+1) overlaps compute(c) ---------------------------
  stage(0, 0);
  WAIT_ASYNCCNT_0();
  __syncthreads();

  for (int c = 0; c < C; ++c) {
    const int buf = c & 1;
    if (c + 1 < C) stage(c + 1, buf ^ 1);

    const float* A = sA[buf];
    const float* Bq = sB[buf];
    if (c < C1) {
      #pragma unroll
      for (int kk = 0; kk < 32; kk += 4) {
        const int r0 = kk + 2 * half;
        v2f a, bv;
        a.x  = A[r0 * 144 + w16 + col];
        a.y  = A[(r0 + 1) * 144 + w16 + col];
        bv.x = Bq[r0 * 16 + col];
        bv.y = Bq[(r0 + 1) * 16 + col];
        acc = __builtin_amdgcn_wmma_f32_16x16x4_f32(
            false, a, false, bv, (short)0, acc, false, false);
      }
    } else {
      #pragma unroll
      for (int kk = 0; kk < 32; kk += 4) {
        const int r0 = kk + 2 * half;
        v2f a = *(const v2f*)&A[(w16 + col) * 36 + r0];   // contiguous K pair
        v2f bv;
        bv.x = Bq[r0 * 16 + col];
        bv.y = Bq[(r0 + 1) * 16 + col];
        acc = __builtin_amdgcn_wmma_f32_16x16x4_f32(
            false, a, false, bv, (short)0, acc, false, false);
      }
    }

    WAIT_ASYNCCNT_0();       // my async stores to LDS for chunk c+1 landed
    __syncthreads();         // everyone done reading buf & done staging buf^1
  }

  // ---- epilogue: + v_in + u broadcast (C/D layout: VGPR r -> rows r / r+8)
  const int  n0   = nb0 + w16;
  const float uval = u_in[b * DD + d0 + col];
  #pragma unroll
  for (int r = 0; r < 8; ++r) {
    const int    n  = n0 + r + 8 * half;
    const size_t ix = ((size_t)b * NN + n) * DD + (d0 + col);
    v_out[ix] = acc[r] + v_in[ix] + uval;
  }
}

// ---------------------------------------------------------------------------
// Streaming pass-through copy (472 MB of tuple outputs): 128-bit non-temporal
// load/store so the edgeAdj stream doesn't evict the L2-resident adj/conEd
// working set the WMMA kernel reuses.
// ---------------------------------------------------------------------------
__global__ __launch_bounds__(256) void copy_nt_f4(const f4* __restrict__ src,
                                                  f4* __restrict__ dst,
                                                  long n4)
{
  long i = (long)blockIdx.x * blockDim.x + threadIdx.x;
  const long stride = (long)gridDim.x * blockDim.x;
  for (; i < n4; i += stride) {
    f4 v = __builtin_nontemporal_load(src + i);
    __builtin_nontemporal_store(v, dst + i);
  }
}

extern "C" void kernel_launch(void* const* d_in, const int* in_sizes, int n_in,
                              void* d_out, int out_size, void* d_ws, size_t ws_size,
                              hipStream_t stream)
{
  const float* v_in    = (const float*)d_in[0];
  const float* e_in    = (const float*)d_in[1];
  const float* u_in    = (const float*)d_in[2];
  const float* adj     = (const float*)d_in[3];
  const float* conEd   = (const float*)d_in[4];
  const float* edgeAdj = (const float*)d_in[5];

  const size_t sz_v    = (size_t)in_sizes[0];
  const size_t sz_e    = (size_t)in_sizes[1];
  const size_t sz_u    = (size_t)in_sizes[2];
  const size_t sz_adj  = (size_t)in_sizes[3];
  const size_t sz_con  = (size_t)in_sizes[4];
  const size_t sz_eadj = (size_t)in_sizes[5];

  float* out      = (float*)d_out;
  float* out_v    = out;
  float* out_e    = out_v   + sz_v;
  float* out_u    = out_e   + sz_e;
  float* out_adj  = out_u   + sz_u;
  float* out_con  = out_adj + sz_adj;
  float* out_eadj = out_con + sz_con;

  auto launch_copy = [&](const float* s, float* d, size_t n) {
    const long n4 = (long)(n / 4);           // all sizes are multiples of 4
    long blocks = (n4 + 255) / 256;
    if (blocks > 8192) blocks = 8192;        // grid-stride handles the rest
    if (blocks < 1) blocks = 1;
    copy_nt_f4<<<dim3((unsigned)blocks), dim3(256), 0, stream>>>(
        (const f4*)s, (f4*)d, n4);
  };

  launch_copy(e_in,    out_e,    sz_e);
  launch_copy(u_in,    out_u,    sz_u);
  launch_copy(adj,     out_adj,  sz_adj);
  launch_copy(conEd,   out_con,  sz_con);
  launch_copy(edgeAdj, out_eadj, sz_eadj);

  // 1024 blocks x 256 threads: b(4) x n-macro(16) x d-tile(16)
  fused_pool_wmma<<<dim3(1024), dim3(256), 0, stream>>>(
      v_in, e_in, u_in, adj, conEd, out_v);
}